// SparseGCNLayer_11811160064780
// MI455X (gfx1250) — compile-verified
//
#include <hip/hip_runtime.h>
#include <hip/hip_bf16.h>

typedef float v2f __attribute__((ext_vector_type(2)));
typedef float v8f __attribute__((ext_vector_type(8)));

#define WAVE_SZ 32
#define GCN_D   512          // fixed by problem (D), power of two
#define NCT     8            // col tiles per wave: D / (4 col-waves * 16)

// ---------------------------------------------------------------------------
// Kernel 1: zero the neigh accumulator (lives in d_ws, poisoned by harness)
// ---------------------------------------------------------------------------
__global__ void gcn_zero_kernel(float4* __restrict__ p, long nvec) {
    long i = (long)blockIdx.x * blockDim.x + threadIdx.x;
    long stride = (long)gridDim.x * blockDim.x;
    float4 z = make_float4(0.f, 0.f, 0.f, 0.f);
    for (; i < nvec; i += stride) p[i] = z;
}

// ---------------------------------------------------------------------------
// Kernel 2: COO SpMM scatter:  neigh[row] += val * x[col]
// One wave cooperatively processes EPW edges; lanes stride the D=512 row with
// float4 loads (coalesced 512B/wave/iter) and global_atomic_add_f32 scatters.
// ---------------------------------------------------------------------------
__global__ __launch_bounds__(256)
void gcn_spmm_kernel(const float* __restrict__ x,
                     const int*   __restrict__ erow,
                     const int*   __restrict__ ecol,
                     const float* __restrict__ eval,
                     float*       __restrict__ neigh,
                     int E, int D, int edges_per_wave) {
    int wave = (blockIdx.x * blockDim.x + threadIdx.x) / WAVE_SZ;
    int lane = threadIdx.x & (WAVE_SZ - 1);
    long e0 = (long)wave * edges_per_wave;
    int nvec = D >> 2;  // float4 chunks per row

    for (int k = 0; k < edges_per_wave; ++k) {
        long e = e0 + k;
        if (e >= E) return;
        int   r = erow[e];
        int   c = ecol[e];
        float v = eval[e];
        const float4* xs = (const float4*)(x + (long)c * D);
        float*        ns = neigh + (long)r * D;
        for (int j = lane; j < nvec; j += WAVE_SZ) {
            float4 g = xs[j];
            atomicAdd(&ns[4 * j + 0], g.x * v);
            atomicAdd(&ns[4 * j + 1], g.y * v);
            atomicAdd(&ns[4 * j + 2], g.z * v);
            atomicAdd(&ns[4 * j + 3], g.w * v);
        }
    }
}

// ---------------------------------------------------------------------------
// Kernel 3: fused dual GEMM + bias + ReLU via V_WMMA_F32_16X16X4_F32.
//   out[n,e] = relu( x @ Wself^T + neigh @ Wneigh^T + bself + bneigh )
//
// Tiling: 256 threads = 8 waves (2 row-groups x 4 col-groups). Each wave owns
// a 16-row x 128-col tile (8 v8f accumulators); the WG covers 32 rows x all
// 512 output columns, so x/neigh stream from HBM exactly once and W stays
// L2-resident.
//
// K-loop is double-buffered: step k+4's 18 fragment loads are issued before
// step k's 16 WMMAs execute, so compute always consumes fragments loaded a
// full compute-phase earlier (loads return in order -> waits resolve with the
// other buffer still outstanding). sched_barrier(0) fences stop the backend
// scheduler from sinking loads back to their consumers (which previously
// coalesced all B fragments onto one VGPR pair and serialized every WMMA).
//
// f32 WMMA fragment layouts (ISA 7.12.2):
//   A (16x4, MxK): lane L holds row M=L&15, K = {0,1}+2*(L>>4)  -> one b64 load
//   B (4x16, KxN): lane L holds col N=L&15, K = {0,1}+2*(L>>4); B[k][n] =
//       W[col0+n][kbase+k] -> same contiguous-in-d b64 load pattern on W.
//   C/D (16x16):   VGPR i, lane L -> row = i + 8*(L>>4), col = L&15
// ---------------------------------------------------------------------------
struct Frags {
    v2f ax, an;
    v2f bs[NCT], bn[NCT];
};

__device__ __forceinline__
Frags load_frags(const float* __restrict__ xrow, const float* __restrict__ nrow,
                 const float* __restrict__ wsrow, const float* __restrict__ wnrow,
                 int k) {
    Frags f;
    f.ax = *(const v2f*)(xrow + k);
    f.an = *(const v2f*)(nrow + k);
    #pragma unroll
    for (int t = 0; t < NCT; ++t) {
        f.bs[t] = *(const v2f*)(wsrow + (long)t * 16 * GCN_D + k);
        f.bn[t] = *(const v2f*)(wnrow + (long)t * 16 * GCN_D + k);
    }
    return f;
}

__device__ __forceinline__
void compute_frags(const Frags& f, v8f (&acc)[NCT]) {
    #pragma unroll
    for (int t = 0; t < NCT; ++t)
        acc[t] = __builtin_amdgcn_wmma_f32_16x16x4_f32(
                     false, f.ax, false, f.bs[t], (short)0, acc[t], false, false);
    #pragma unroll
    for (int t = 0; t < NCT; ++t)
        acc[t] = __builtin_amdgcn_wmma_f32_16x16x4_f32(
                     false, f.an, false, f.bn[t], (short)0, acc[t], false, false);
}

__global__ __launch_bounds__(256, 1)
void gcn_gemm_kernel(const float* __restrict__ x,
                     const float* __restrict__ neigh,
                     const float* __restrict__ Wself,
                     const float* __restrict__ Wneigh,
                     const float* __restrict__ bself,
                     const float* __restrict__ bneigh,
                     float*       __restrict__ out,
                     int N) {
    const int D = GCN_D;

    int wid  = threadIdx.x >> 5;
    int lane = threadIdx.x & 31;
    int rw = wid >> 2;                    // 0..1  row-wave within WG
    int cw = wid & 3;                     // 0..3  col-wave within WG

    int row0 = (blockIdx.x * 2 + rw) * 16;     // node-tile base (wave uniform)
    int col0 = cw * (NCT * 16);                // 128-col slab base
    if (row0 >= N) return;                      // wave-uniform exit

    int lm = lane & 15;   // fragment row/col within 16x16 tile
    int lh = lane >> 4;   // K sub-offset selector

    int arow = row0 + lm;
    if (arow >= N) arow = N - 1;                // clamp tail reads (stores guarded)
    int koff = lh * 2;

    const float* xrow  = x      + (long)arow * D + koff;
    const float* nrow  = neigh  + (long)arow * D + koff;
    const float* wsrow = Wself  + (long)(col0 + lm) * D + koff;
    const float* wnrow = Wneigh + (long)(col0 + lm) * D + koff;

    v8f acc[NCT];
    #pragma unroll
    for (int t = 0; t < NCT; ++t) acc[t] = v8f{};

    // Software-pipelined, double-buffered K loop (D/8 iterations, 32 WMMA each)
    Frags f0 = load_frags(xrow, nrow, wsrow, wnrow, 0);
    #pragma unroll 1
    for (int k = 0; k < D; k += 8) {
        Frags f1 = load_frags(xrow, nrow, wsrow, wnrow, k + 4);
        __builtin_amdgcn_sched_barrier(0);
        compute_frags(f0, acc);                       // f0 landed last phase
        __builtin_amdgcn_sched_barrier(0);
        f0 = load_frags(xrow, nrow, wsrow, wnrow, (k + 8) & (D - 1));
        __builtin_amdgcn_sched_barrier(0);
        compute_frags(f1, acc);                       // f1 overlapped compute(f0)
        __builtin_amdgcn_sched_barrier(0);
    }

    // Epilogue: bias + ReLU + store (C/D layout per ISA 7.12.2)
    int orow0 = row0 + lh * 8;
    #pragma unroll
    for (int t = 0; t < NCT; ++t) {
        int ocol = col0 + t * 16 + lm;
        float bias = bself[ocol] + bneigh[ocol];
        #pragma unroll
        for (int i = 0; i < 8; ++i) {
            int r = orow0 + i;
            if (r < N) {
                float v = acc[t][i] + bias;
                out[(long)r * D + ocol] = v > 0.f ? v : 0.f;
            }
        }
    }
}

// ---------------------------------------------------------------------------
extern "C" void kernel_launch(void* const* d_in, const int* in_sizes, int n_in,
                              void* d_out, int out_size, void* d_ws, size_t ws_size,
                              hipStream_t stream) {
    const float* x      = (const float*)d_in[0];
    const int*   erow   = (const int*)  d_in[1];
    const int*   ecol   = (const int*)  d_in[2];
    const float* eval   = (const float*)d_in[3];
    const float* Wself  = (const float*)d_in[4];
    const float* bself  = (const float*)d_in[5];
    const float* Wneigh = (const float*)d_in[6];
    const float* bneigh = (const float*)d_in[7];
    float* out = (float*)d_out;

    int D = in_sizes[5];            // b_self has D elements (== GCN_D == 512)
    int N = in_sizes[0] / D;        // x is N*D
    int E = in_sizes[1];            // edge_row has E elements

    float* neigh = (float*)d_ws;    // N*D fp32 scratch accumulator

    // 1) zero neigh
    long nvec = ((long)N * D) >> 2;
    gcn_zero_kernel<<<4096, 256, 0, stream>>>((float4*)neigh, nvec);

    // 2) SpMM scatter
    const int EPW = 8;                                   // edges per wave
    long waves  = (E + EPW - 1) / EPW;
    long blocks = (waves + 7) / 8;                       // 8 waves / 256-thr WG
    gcn_spmm_kernel<<<(int)blocks, 256, 0, stream>>>(x, erow, ecol, eval,
                                                     neigh, E, D, EPW);

    // 3) fused WMMA GEMMs + bias + ReLU: WG covers 32 rows x all 512 cols
    dim3 grid((N + 31) / 32);
    gcn_gemm_kernel<<<grid, 256, 0, stream>>>(x, neigh, Wself, Wneigh,
                                              bself, bneigh, out, N);
}